// RoiPooling_11450382811677
// MI455X (gfx1250) — compile-verified
//
#include <hip/hip_runtime.h>
#include <stdint.h>

#define PH 7
#define PW 7
#define FH 256   // feature-map height
#define FW 256   // feature-map width
#define FC 256   // channels

typedef __attribute__((ext_vector_type(4))) float f4;

// Explicit CDNA5 dependency-counter waits (split counters, gfx1250).
#define WAIT_ASYNCCNT(n) asm volatile("s_wait_asynccnt " n ::: "memory")
#define WAIT_DSCNT0()    asm volatile("s_wait_dscnt 0x0" ::: "memory")

__device__ __forceinline__ f4 vmax4(f4 a, f4 b) {
  f4 r;
  r.x = __builtin_fmaxf(a.x, b.x);
  r.y = __builtin_fmaxf(a.y, b.y);
  r.z = __builtin_fmaxf(a.z, b.z);
  r.w = __builtin_fmaxf(a.w, b.w);
  return r;
}

// Async DMA of 16 bytes/lane: global -> LDS (CDNA5 GLOBAL_LOAD_ASYNC_TO_LDS_B128).
// Inline asm (GV mode): vdst = LDS byte offset in a VGPR, vaddr = 64-bit global
// address VGPR pair, SADDR = off. Low 32 bits of a generic shared pointer are
// the LDS byte offset (ISA aperture layout). Completion tracked via ASYNCcnt.
__device__ __forceinline__ void async_cp16(const float* g, f4* l) {
  uint32_t lds_off = (uint32_t)(uintptr_t)l;
  asm volatile("global_load_async_to_lds_b128 %0, %1, off"
               :: "v"(lds_off), "v"(g) : "memory");
}

__global__ __launch_bounds__(448)
void roi_pool_kernel(const float* __restrict__ x,
                     const float* __restrict__ rois,
                     float* __restrict__ out, int N)
{
  __shared__ f4 buf[4][PW * 64];   // 4-deep ring of 7KB row groups (28 KB)

  const int i    = blockIdx.x;     // pooled row bin 0..6
  const int n    = blockIdx.y;     // roi index
  const int c4   = threadIdx.x;    // float4 channel group 0..63
  const int j    = threadIdx.y;    // pooled col bin 0..6
  const int slot = j * 64 + c4;    // thread-private LDS slot within a buffer

  // Decode roi: values are exact integers scaled by 256.
  const float inv = 1.0f / 256.0f;
  const int r0 = (int)(rois[n * 4 + 0] * inv);
  const int r1 = (int)(rois[n * 4 + 1] * inv);
  const int r2 = (int)(rois[n * 4 + 2] * inv);
  const int r3 = (int)(rois[n * 4 + 3] * inv);
  const int wst = (r2 - r0) / PW;
  const int hst = (r3 - r1) / PH;

  const int start = r0 + i * hst;          // quirk: x-coord used as row start
  const int nrows = i * hst;               // row range [start, start + i*hst)

  float* op = out + ((((size_t)n * PH + i) * PW + j) * FC + c4 * 4);

  const float ninf = -__builtin_inff();
  f4 acc = {ninf, ninf, ninf, ninf};

  if (nrows <= 0) {                        // i == 0: empty reduction -> -inf
    *(f4*)op = acc;
    return;
  }

  const int col = start + (j + 1) * wst;   // quirk: columns start at (j+1)*w
  const float* g = x + (((size_t)start * FW + col) * FC + c4 * 4);
  const size_t rs = (size_t)FW * FC;       // row stride in floats

  if (nrows < 4) {                         // defensive fallback (unreached for given ranges)
    for (int r = 0; r < nrows; ++r)
      acc = vmax4(acc, *(const f4*)(g + (size_t)r * rs));
    *(f4*)op = acc;
    return;
  }

  // Prologue: 3 rows in flight.
  async_cp16(g + 0 * rs, &buf[0][slot]);
  async_cp16(g + 1 * rs, &buf[1][slot]);
  async_cp16(g + 2 * rs, &buf[2][slot]);

  int idx = 0;
  // Steady state: issue row idx+3, wait for row idx, consume it.
  for (; idx < nrows - 3; ++idx) {
    async_cp16(g + (size_t)(idx + 3) * rs, &buf[(idx + 3) & 3][slot]);
    WAIT_ASYNCCNT("0x3");                  // oldest (row idx) has landed in LDS
    f4 v = buf[idx & 3][slot];             // ds_load_b128 from private slot
    acc = vmax4(acc, v);
    WAIT_DSCNT0();                         // slot free before it's re-targeted next iter
  }
  // Tail: drain remaining 3 in-flight rows.
  WAIT_ASYNCCNT("0x2");
  acc = vmax4(acc, buf[idx & 3][slot]); WAIT_DSCNT0(); ++idx;
  WAIT_ASYNCCNT("0x1");
  acc = vmax4(acc, buf[idx & 3][slot]); WAIT_DSCNT0(); ++idx;
  WAIT_ASYNCCNT("0x0");
  acc = vmax4(acc, buf[idx & 3][slot]);

  *(f4*)op = acc;                          // coalesced b128 store
}

extern "C" void kernel_launch(void* const* d_in, const int* in_sizes, int n_in,
                              void* d_out, int out_size, void* d_ws, size_t ws_size,
                              hipStream_t stream) {
  const float* x    = (const float*)d_in[0];   // [1, 256, 256, 256] f32
  const float* rois = (const float*)d_in[1];   // [1, N, 4] f32
  float* out = (float*)d_out;                  // [1, N, 7, 7, 256] f32
  const int N = in_sizes[1] / 4;

  dim3 grid(PH, N);        // one block per (row-bin, roi)
  dim3 block(64, PW);      // (float4-channel-group, col-bin) = 448 threads = 14 waves
  roi_pool_kernel<<<grid, block, 0, stream>>>(x, rois, out, N);
}